// SparseNeuSRenderer_32186484916659
// MI455X (gfx1250) — compile-verified
//
#include <hip/hip_runtime.h>
#include <cstdint>
#include <cstddef>

// ---------------------------------------------------------------------------
// NeuS up-sampling pass for MI455X (gfx1250).
//
// Memory-bound workload (~300 MB of streamed traffic -> ~13 us floor at
// 23.3 TB/s).  No GEMM structure, so no WMMA; the CDNA5 feature that pays is
// the async global->LDS DMA path: the 96^3 occupancy volume is packed into a
// 108 KB bitmask which every block stages into LDS with
// global_load_async_to_lds_b128 (tracked by ASYNCcnt / s_wait_asynccnt),
// turning 16.7M random 4B global gathers into conflict-free ds_load bit tests.
// ---------------------------------------------------------------------------

#define N_RAYS     131072
#define NS         64
#define NI         64
#define NALL       (NS + NI)                 // 128
#define VOLD       96
#define MASK_WORDS ((VOLD * VOLD * VOLD) / 32)  // 27648 dwords
#define MASK_BYTES (MASK_WORDS * 4)             // 110592 B  (108 KB)
#define BLOCK      128
#define WBUF_BYTES ((NS) * BLOCK * 4)           // 32 KB weight scratch
#define SMEM_BYTES (MASK_BYTES + WBUF_BYTES)    // 143360 B -> 2 blocks / WGP

// ---------------- kernel 1: pack float occupancy volume into a bitmask ----
__global__ void pack_mask_kernel(const float* __restrict__ vol,
                                 unsigned* __restrict__ bits) {
  int w = blockIdx.x * blockDim.x + threadIdx.x;
  if (w >= MASK_WORDS) return;
  const float* p = vol + (size_t)w * 32;   // 128B contiguous per thread
  unsigned v = 0u;
#pragma unroll
  for (int b = 0; b < 32; ++b)
    v |= (p[b] > 0.0f ? 1u : 0u) << b;
  bits[w] = v;
}

// idx formula from the reference collapses exactly:
//   floor(((c+1)*96 - 1)*0.5 + 0.5) == floor((c+1)*48)
__device__ __forceinline__ float mask_lookup_lds(const unsigned* __restrict__ mlds,
                                                 float x, float y, float z) {
  int ix = (int)floorf((x + 1.0f) * 48.0f);
  int iy = (int)floorf((y + 1.0f) * 48.0f);
  int iz = (int)floorf((z + 1.0f) * 48.0f);
  if (((unsigned)ix >= VOLD) | ((unsigned)iy >= VOLD) | ((unsigned)iz >= VOLD))
    return 0.0f;                      // out-of-bounds -> 0, same as reference
  int L = (ix * VOLD + iy) * VOLD + iz;
  return (float)((mlds[L >> 5] >> (L & 31)) & 1u);
}

extern __shared__ unsigned char smem_raw[];

// ---------------- kernel 2: fused alpha / inverse-CDF sample / merge ------
__global__ void __launch_bounds__(BLOCK)
neus_upsample_kernel(const float* __restrict__ rays_o,
                     const float* __restrict__ rays_d,
                     const float* __restrict__ z_vals,
                     const float* __restrict__ sdf,
                     const float* __restrict__ new_sdf_raw,
                     const float* __restrict__ inv_variance,
                     const unsigned* __restrict__ mask_bits,
                     float* __restrict__ out_zsamp,
                     float* __restrict__ out_zsort,
                     float* __restrict__ out_ssort) {
  unsigned* mlds = (unsigned*)smem_raw;                 // 108 KB bitmask
  float*    wbuf = (float*)(smem_raw + MASK_BYTES);     // per-thread weights
  const int t = threadIdx.x;

  // --- Stage occupancy bitmask into LDS via CDNA5 async global->LDS DMA ---
  {
    unsigned ldsbase = __builtin_amdgcn_groupstaticsize();  // dynamic-LDS start
    const int chunks = MASK_BYTES / 16;                     // 6912 b128 moves
    for (int c = t; c < chunks; c += BLOCK) {               // 54 per wave < 63
      unsigned ldsoff = ldsbase + ((unsigned)c << 4);
      const unsigned char* src = (const unsigned char*)mask_bits + ((size_t)c << 4);
      asm volatile("global_load_async_to_lds_b128 %0, %1, off"
                   :: "v"(ldsoff), "v"(src)
                   : "memory");
    }
    asm volatile("s_wait_asynccnt 0" ::: "memory");
  }
  __syncthreads();

  const int r = blockIdx.x * BLOCK + t;                 // one ray per thread
  const float ox = rays_o[r * 3 + 0], oy = rays_o[r * 3 + 1], oz = rays_o[r * 3 + 2];
  const float dx = rays_d[r * 3 + 0], dy = rays_d[r * 3 + 1], dz = rays_d[r * 3 + 2];
  const float inv_s = inv_variance[0];
  const float* zrow = z_vals + (size_t)r * NS;
  const float* srow = sdf    + (size_t)r * NS;

  // ---- pass 1: alpha / transmittance / per-interval weights -> LDS ------
  float pz = zrow[0], ps = srow[0];
  float pm_prev = mask_lookup_lds(mlds, ox + dx * pz, oy + dy * pz, oz + dz * pz);
  float prev_raw = 0.0f;               // prev_dot[0] == 0 in the reference
  float T = 1.0f, wsum = 0.0f;
  for (int i = 0; i < NS - 1; ++i) {
    float nz = zrow[i + 1], nsd = srow[i + 1];
    float pm_next = mask_lookup_lds(mlds, ox + dx * nz, oy + dy * nz, oz + dz * nz);
    float ptsm = pm_prev * pm_next;
    float mid  = 0.5f * (ps + nsd);
    float dzv  = nz - pz;
    float raw  = (nsd - ps) / (dzv + 1e-5f);
    float dot  = fminf(prev_raw, raw);
    dot = fminf(fmaxf(dot, -10.0f), 0.0f) * ptsm;
    float pe = mid - 0.5f * dot * dzv;
    float ne = mid + 0.5f * dot * dzv;
    float pc = 1.0f / (1.0f + __expf(-pe * inv_s));
    float nc = 1.0f / (1.0f + __expf(-ne * inv_s));
    float alpha = ptsm * ((pc - nc + 1e-5f) / (pc + 1e-5f));
    float w = alpha * T + 1e-5f;       // weights + 1e-5 (pdf regularizer)
    T *= (1.0f - alpha + 1e-7f);
    wbuf[i * BLOCK + t] = w;           // bank-conflict-free layout
    wsum += w;
    prev_raw = raw; pm_prev = pm_next; pz = nz; ps = nsd;
  }

  // ---- pass 2: monotone inverse-CDF sampling (two-pointer, O(S+NI)) -----
  const float inv_wsum = 1.0f / wsum;
  float* zsamp = out_zsamp + (size_t)r * NI;
  float* ssort = out_ssort + (size_t)r * NALL;
  float* stage = ssort + NS;           // stage new_sdf in upper half of output
  int   j = 0;
  float cdf_b = 0.0f;
  float cdf_a = wbuf[t] * inv_wsum;    // cdf[1]
  float zb = zrow[0], za = zrow[1];
  for (int k = 0; k < NI; ++k) {
    float u = (k + 0.5f) * (1.0f / NI);              // linspace(.5/NI, 1-.5/NI)
    while (j < NS - 2 && cdf_a <= u) {               // searchsorted 'right'
      ++j;
      cdf_b = cdf_a;
      cdf_a += wbuf[j * BLOCK + t] * inv_wsum;
      zb = za;
      za = zrow[j + 1];
    }
    float denom = cdf_a - cdf_b;
    denom = (denom < 1e-5f) ? 1.0f : denom;
    float tt = (u - cdf_b) / denom;
    float zs = zb + tt * (za - zb);
    zsamp[k] = zs;                                   // output 1 (also reused)
    float m = mask_lookup_lds(mlds, ox + dx * zs, oy + dy * zs, oz + dz * zs);
    stage[k] = (m > 0.0f) ? new_sdf_raw[(size_t)r * NI + k] : 100.0f;
  }

  // ---- pass 3: stable 64+64 merge (z_samples is sorted by construction) --
  float* zsort = out_zsort + (size_t)r * NALL;
  int a = 0, b = 0;
  float va = zrow[0],  sa = srow[0];
  float vb = zsamp[0], sb = stage[0];
  for (int o = 0; o < NALL; ++o) {
    bool takeA = (b >= NI) || ((a < NS) && (va <= vb));  // ties: z_vals first
    if (takeA) {
      zsort[o] = va; ssort[o] = sa;
      ++a;
      if (a < NS) { va = zrow[a]; sa = srow[a]; }
    } else {
      zsort[o] = vb; ssort[o] = sb;   // never clobbers unread stage slots
      ++b;
      if (b < NI) { vb = zsamp[b]; sb = stage[b]; }
    }
  }
}

// ---------------------------------------------------------------------------
extern "C" void kernel_launch(void* const* d_in, const int* in_sizes, int n_in,
                              void* d_out, int out_size, void* d_ws, size_t ws_size,
                              hipStream_t stream) {
  const float* rays_o      = (const float*)d_in[0];
  const float* rays_d      = (const float*)d_in[1];
  const float* z_vals      = (const float*)d_in[2];
  const float* sdf         = (const float*)d_in[3];
  const float* new_sdf_raw = (const float*)d_in[4];
  const float* inv_var     = (const float*)d_in[5];
  const float* mask_volume = (const float*)d_in[6];

  unsigned* bits = (unsigned*)d_ws;          // needs 110,592 B of scratch
  float* out        = (float*)d_out;
  float* out_zsamp  = out;                                   // [N_RAYS, 64]
  float* out_zsort  = out + (size_t)N_RAYS * NI;             // [N_RAYS, 128]
  float* out_ssort  = out_zsort + (size_t)N_RAYS * NALL;     // [N_RAYS, 128]

  // 1) pack 3.5 MB float volume -> 108 KB bitmask (32:1 compression)
  pack_mask_kernel<<<MASK_WORDS / 256, 256, 0, stream>>>(mask_volume, bits);

  // allow 140 KB of dynamic LDS (gfx1250 WGP has 320 KB)
  (void)hipFuncSetAttribute((const void*)neus_upsample_kernel,
                            hipFuncAttributeMaxDynamicSharedMemorySize,
                            SMEM_BYTES);

  // 2) fused alpha -> cdf -> importance sample -> merge, one ray per thread
  neus_upsample_kernel<<<N_RAYS / BLOCK, BLOCK, SMEM_BYTES, stream>>>(
      rays_o, rays_d, z_vals, sdf, new_sdf_raw, inv_var, bits,
      out_zsamp, out_zsort, out_ssort);
}